// SOM_45810121179831
// MI455X (gfx1250) — compile-verified
//
#include <hip/hip_runtime.h>
#include <hip/hip_bf16.h>

// ---------------------------------------------------------------------------
// SOM update for MI455X (gfx1250, wave32).
//   Phase 1: BMU search via V_WMMA_F32_16X16X4_F32 (W streamed through A,
//            x broadcast through B), packed-u64 atomic argmin.
//   Phase 2: fused elementwise neighborhood update (memory-bound, float4).
// ---------------------------------------------------------------------------

typedef __attribute__((ext_vector_type(2))) float v2f;
typedef __attribute__((ext_vector_type(8))) float v8f;

#define SOM_DIM   256
#define SOM_M     512
#define SOM_N     512
#define SOM_ROWS  (SOM_M * SOM_N)
#define SOM_NITER 100.0f
#define SOM_ALPHA 0.3f
#define SOM_SIGMA 256.0f   // max(M,N)/2

// ---------------------------------------------------------------------------
// Kernel 0: reset the packed argmin key (must run every call: deterministic).
// ---------------------------------------------------------------------------
__global__ void som_init_kernel(unsigned long long* best) {
    *best = ~0ull;
}

// ---------------------------------------------------------------------------
// Kernel 1: BMU search. One wave (32 lanes) owns a 16-row tile.
//
// For each K-chunk of 4 (64 chunks over DIM=256):
//   A (16x4 f32, ISA layout): lane L holds W[rowBase+(L&15)][cb+(L>>4)*2+{0,1}]
//   B (4x16 f32):             lane L holds x[cb+(L>>4)*2+{0,1}] (bcast cols)
//   C += A*B  ->  after 64 chunks, D[m][*] = dot(W[rowBase+m], x) in every col.
// Row norms accumulated from the same A registers; dist^2 ranking uses
// ||w||^2 - 2*dot (||x||^2 is a constant offset, dropped).
// ---------------------------------------------------------------------------
__global__ __launch_bounds__(256) void som_bmu_kernel(
    const float* __restrict__ x,
    const float* __restrict__ w,
    unsigned long long* __restrict__ best)
{
    __shared__ float xs[SOM_DIM];
    const int tid = threadIdx.x;
    xs[tid] = x[tid];                       // blockDim.x == SOM_DIM == 256
    __syncthreads();

    const int lane    = tid & 31;
    const int wave    = tid >> 5;                       // 0..7
    const int tile    = blockIdx.x * 8 + wave;          // 16-row tile
    const int rowBase = tile * 16;
    const int half    = lane >> 4;                      // 0 or 1
    const int r       = lane & 15;                      // A-matrix row M
    const int koff    = half * 2;                       // per-lane k offset

    const float* wrow = w + (size_t)(rowBase + r) * SOM_DIM + koff;

    v8f  c   = {0.f, 0.f, 0.f, 0.f, 0.f, 0.f, 0.f, 0.f};
    float nrm = 0.f;

#pragma unroll 4
    for (int cb = 0; cb < SOM_DIM; cb += 4) {
        v2f a, b;
        a.x = wrow[cb];
        a.y = wrow[cb + 1];
        b.x = xs[cb + koff];
        b.y = xs[cb + koff + 1];
        // D = A x B + C : emits v_wmma_f32_16x16x4_f32
        c = __builtin_amdgcn_wmma_f32_16x16x4_f32(
                /*neg_a=*/false, a, /*neg_b=*/false, b,
                /*c_mod=*/(short)0, c, /*reuse_a=*/false, /*reuse_b=*/false);
        nrm = fmaf(a.x, a.x, nrm);
        nrm = fmaf(a.y, a.y, nrm);
    }

    // Lanes r and r^16 each hold half of ||w_{rowBase+r}||^2 -> combine.
    nrm += __shfl_xor(nrm, 16, 32);         // full row norm in lanes r, r+16

    // C/D layout: lanes 0-15 hold M=0..7 in c[0..7]; lanes 16-31 hold M=8..15.
    const int mbase = half * 8;
    unsigned long long bestKey = ~0ull;
#pragma unroll
    for (int v = 0; v < 8; ++v) {
        const int   m   = mbase + v;
        const float dot = c[v];
        const float nr  = __shfl(nrm, m, 32);   // row norm lives in lane m
        const float d   = nr - 2.0f * dot;      // dist^2 minus const ||x||^2
        unsigned ub = __float_as_uint(d);
        ub = (ub & 0x80000000u) ? ~ub : (ub | 0x80000000u); // order-preserving
        const unsigned long long key =
            ((unsigned long long)ub << 32) | (unsigned)(rowBase + m);
        bestKey = (key < bestKey) ? key : bestKey;
    }

    // Wave32 butterfly min-reduce.
#pragma unroll
    for (int off = 16; off >= 1; off >>= 1) {
        const unsigned long long o = __shfl_xor(bestKey, off, 32);
        bestKey = (o < bestKey) ? o : bestKey;
    }
    if (lane == 0) atomicMin(best, bestKey);    // global_atomic_min_u64
}

// ---------------------------------------------------------------------------
// Kernel 2: fused neighborhood update, one float4 per thread.
// new_w = w + alpha_op * exp(-d2/sigma_op^2) * (x - w)
// ---------------------------------------------------------------------------
__global__ __launch_bounds__(256) void som_update_kernel(
    const float* __restrict__ x,
    const float* __restrict__ w,
    const int* __restrict__ itp,
    const unsigned long long* __restrict__ best,
    float* __restrict__ out)
{
    const size_t gid = (size_t)blockIdx.x * blockDim.x + threadIdx.x;
    const size_t e   = gid * 4;                  // element offset
    const int row = (int)(e >> 8);               // / SOM_DIM
    const int col = (int)(e & (SOM_DIM - 1));

    const unsigned bidx = (unsigned)(*best & 0xFFFFFFFFu);
    const int bi = (int)(bidx >> 9);             // / SOM_N
    const int bj = (int)(bidx & (SOM_N - 1));

    const float lr       = 1.0f - (float)(*itp) / SOM_NITER;
    const float alpha_op = SOM_ALPHA * lr;
    const float sigma_op = SOM_SIGMA * lr;

    const int i = row >> 9;                      // / SOM_N
    const int j = row & (SOM_N - 1);
    const float di = (float)(i - bi);
    const float dj = (float)(j - bj);
    const float d2 = di * di + dj * dj;
    const float mult = alpha_op * __expf(-d2 / (sigma_op * sigma_op));

    const float4 w4 = *(const float4*)(w + e);
    const float4 x4 = *(const float4*)(x + col);
    float4 o;
    o.x = fmaf(mult, x4.x - w4.x, w4.x);
    o.y = fmaf(mult, x4.y - w4.y, w4.y);
    o.z = fmaf(mult, x4.z - w4.z, w4.z);
    o.w = fmaf(mult, x4.w - w4.w, w4.w);
    *(float4*)(out + e) = o;
}

// ---------------------------------------------------------------------------
extern "C" void kernel_launch(void* const* d_in, const int* in_sizes, int n_in,
                              void* d_out, int out_size, void* d_ws, size_t ws_size,
                              hipStream_t stream) {
    const float* x = (const float*)d_in[0];   // [256]
    const float* w = (const float*)d_in[1];   // [262144, 256]
    const int*  it = (const int*)d_in[2];     // scalar
    float* out     = (float*)d_out;           // [262144, 256]
    unsigned long long* best = (unsigned long long*)d_ws;

    (void)in_sizes; (void)n_in; (void)out_size; (void)ws_size;

    som_init_kernel<<<1, 1, 0, stream>>>(best);

    // 262144 rows / 16 rows-per-wave / 8 waves-per-block = 2048 blocks
    som_bmu_kernel<<<SOM_ROWS / (16 * 8), 256, 0, stream>>>(x, w, best);

    // 67108864 elements / 4 per thread / 256 threads = 65536 blocks
    som_update_kernel<<<(SOM_ROWS * (SOM_DIM / 4)) / 256, 256, 0, stream>>>(
        x, w, it, best, out);
}